// MG_13408887898961
// MI455X (gfx1250) — compile-verified
//
#include <hip/hip_runtime.h>
#include <math.h>

#define IN_H  128
#define OUT_H 256
#define BN_EPS 1e-5f
#define NORM_EPS 1e-12f

typedef __attribute__((ext_vector_type(2)))  float  v2f;
typedef __attribute__((ext_vector_type(8)))  float  v8f;

// ---------------- degree / norm / mask ----------------

__global__ void k_deg(const int* __restrict__ src, const int* __restrict__ dst,
                      float* degO, float* degI, int E) {
  int e = blockIdx.x * blockDim.x + threadIdx.x;
  if (e < E) {
    atomicAdd(&degO[src[e]], 1.0f);   // integer-valued adds: order-independent, exact
    atomicAdd(&degI[dst[e]], 1.0f);
  }
}

__global__ void k_norm_inplace(float* degO, float* degI, int n) {
  int i = blockIdx.x * blockDim.x + threadIdx.x;
  if (i < n) {
    degO[i] = rsqrtf(fmaxf(degO[i], 1.0f));
    degI[i] = rsqrtf(fmaxf(degI[i], 1.0f));
  }
}

__global__ void k_mark(const int* __restrict__ ids, int m, int* __restrict__ flag) {
  int i = blockIdx.x * blockDim.x + threadIdx.x;
  if (i < m) flag[ids[i]] = 1;
}

// ---------------- fused 128-wide SpMM (3 outputs share one gather) ----------------
// one wave (32 lanes) per edge, each lane owns 4 features (float4)

__global__ void k_spmm128_fused(const float4* __restrict__ feat,
                                const float4* __restrict__ tok0,
                                const float4* __restrict__ tok1,
                                const int* __restrict__ src, const int* __restrict__ dst,
                                const float* __restrict__ ns,
                                const int* __restrict__ m0, const int* __restrict__ m1,
                                float* __restrict__ aggF, float* __restrict__ aggX,
                                float* __restrict__ aggX1, int E) {
  long long t = (long long)blockIdx.x * blockDim.x + threadIdx.x;
  int e = (int)(t >> 5);
  int lane = (int)(t & 31);
  if (e >= E) return;
  int s = src[e], d = dst[e];
  float n = ns[s];
  float4 f = feat[(size_t)s * 32 + lane];
  float vx = f.x * n, vy = f.y * n, vz = f.z * n, vw = f.w * n;
  float ax = vx, ay = vy, az = vz, aw = vw;       // branch-1 encoder input (mask0)
  float bx = vx, by = vy, bz = vz, bw = vw;       // branch-2 decoder input (mask1)
  if (m0[s]) { float4 t0 = tok0[lane]; ax = t0.x*n; ay = t0.y*n; az = t0.z*n; aw = t0.w*n; }
  if (m1[s]) { float4 t1 = tok1[lane]; bx = t1.x*n; by = t1.y*n; bz = t1.z*n; bw = t1.w*n; }
  size_t base = (size_t)d * 128 + lane * 4;
  atomicAdd(&aggF [base+0], vx); atomicAdd(&aggF [base+1], vy);
  atomicAdd(&aggF [base+2], vz); atomicAdd(&aggF [base+3], vw);
  atomicAdd(&aggX [base+0], ax); atomicAdd(&aggX [base+1], ay);
  atomicAdd(&aggX [base+2], az); atomicAdd(&aggX [base+3], aw);
  atomicAdd(&aggX1[base+0], bx); atomicAdd(&aggX1[base+1], by);
  atomicAdd(&aggX1[base+2], bz); atomicAdd(&aggX1[base+3], bw);
}

// ---------------- 256-wide SpMM (layer 1 of encoders) ----------------

__global__ void k_spmm256(const float4* __restrict__ H,
                          const int* __restrict__ src, const int* __restrict__ dst,
                          const float* __restrict__ ns,
                          float* __restrict__ agg, int E) {
  long long t = (long long)blockIdx.x * blockDim.x + threadIdx.x;
  int e = (int)(t >> 5);
  int lane = (int)(t & 31);
  if (e >= E) return;
  int s = src[e], d = dst[e];
  float n = ns[s];
#pragma unroll
  for (int j = 0; j < 2; j++) {
    float4 h = H[(size_t)s * 64 + j * 32 + lane];
    size_t base = (size_t)d * 256 + j * 128 + lane * 4;
    atomicAdd(&agg[base+0], h.x*n); atomicAdd(&agg[base+1], h.y*n);
    atomicAdd(&agg[base+2], h.z*n); atomicAdd(&agg[base+3], h.w*n);
  }
}

// ---------------- WMMA GEMM: Y = PReLU((A*nd) @ W + b, ca) ----------------
// A: [n,K] row-major, W: [K,256] row-major.
// One wave computes a 16x32 strip: one A fragment feeds two 16x16 WMMA tiles.
// A-frag (16x4 f32): lanes 0-15 M=lane K={k,k+1}; lanes 16-31 same M, K={k+2,k+3}.
// B-frag mirrors A with N in place of M. C/D: VGPR v -> M=v (+8 for hi half), N=lane&15.
// Row index is CLAMPED (not predicated) for loads so EXEC stays uniform in the
// K-loop (WMMA needs EXEC=all-ones; predication would also serialize the loads).

__global__ __launch_bounds__(256)
void k_gemm_bias_prelu(const float* __restrict__ A, const float* __restrict__ nd,
                       const float* __restrict__ W, const float* __restrict__ bias,
                       const float* __restrict__ ca, float* __restrict__ Y,
                       int n, int K) {
  const int lane = threadIdx.x;                 // 0..31
  const int wave = threadIdx.y;                 // 0..7
  const int rowbase = blockIdx.x * 16;
  const int colbase = wave * 32;                // 8 waves x 32 cols = 256
  const int m    = lane & 15;
  const int half = lane >> 4;
  int row = rowbase + m;
  if (row >= n) row = n - 1;                    // clamp: loads valid, stores guarded
  const float scale = nd[row];
  const float* __restrict__ Arow = A + (size_t)row * K;
  const int koff = half * 2;
  v8f c0 = {}, c1 = {};
  for (int k = 0; k < K; k += 4) {
    float2 av = *(const float2*)(Arow + k + koff);
    v2f a;
    a.x = av.x * scale;
    a.y = av.y * scale;
    const float* __restrict__ Wr0 = W + (size_t)(k + koff)     * OUT_H + colbase + m;
    const float* __restrict__ Wr1 = W + (size_t)(k + koff + 1) * OUT_H + colbase + m;
    v2f b0, b1;
    b0.x = Wr0[0];  b0.y = Wr1[0];
    b1.x = Wr0[16]; b1.y = Wr1[16];
    c0 = __builtin_amdgcn_wmma_f32_16x16x4_f32(false, a, false, b0, (short)0, c0,
                                               false, false);
    c1 = __builtin_amdgcn_wmma_f32_16x16x4_f32(false, a, false, b1, (short)0, c1,
                                               false, false);
  }
  const float cb = ca[0];
#pragma unroll
  for (int v = 0; v < 8; v++) {
    int r = rowbase + v + half * 8;
    if (r < n) {
      int col = colbase + m;
      float z0 = c0[v] + bias[col];
      z0 = (z0 >= 0.0f) ? z0 : cb * z0;
      Y[(size_t)r * OUT_H + col] = z0;
      float z1 = c1[v] + bias[col + 16];
      z1 = (z1 >= 0.0f) ? z1 : cb * z1;
      Y[(size_t)r * OUT_H + col + 16] = z1;
    }
  }
}

// ---------------- BatchNorm (training stats over nodes) ----------------

__global__ void k_bn_stats(const float* __restrict__ Y, float* __restrict__ bnS,
                           float* __restrict__ bnQ, int n) {
  int c = threadIdx.x;                    // 256 columns
  int r0 = blockIdx.x * 64;
  int r1 = r0 + 64; if (r1 > n) r1 = n;
  float s = 0.0f, q = 0.0f;
  for (int r = r0; r < r1; r++) {
    float y = Y[(size_t)r * OUT_H + c];
    s += y; q += y * y;
  }
  atomicAdd(&bnS[c], s);
  atomicAdd(&bnQ[c], q);
}

__global__ void k_bn_apply(float* __restrict__ Y, const float* __restrict__ bnS,
                           const float* __restrict__ bnQ,
                           const float* __restrict__ g, const float* __restrict__ be,
                           const float* __restrict__ aa, int n) {
  size_t i = (size_t)blockIdx.x * blockDim.x + threadIdx.x;
  if (i >= (size_t)n * OUT_H) return;
  int c = (int)(i & (OUT_H - 1));
  float invN = 1.0f / (float)n;
  float mu = bnS[c] * invN;
  float var = fmaxf(bnQ[c] * invN - mu * mu, 0.0f);
  float y = (Y[i] - mu) * rsqrtf(var + BN_EPS) * g[c] + be[c];
  float a = aa[0];
  Y[i] = (y >= 0.0f) ? y : a * y;
}

// ---------------- SCE loss over masked rows (wave per node) ----------------

__global__ void k_sce(const float* __restrict__ X, const float* __restrict__ Yc,
                      const int* __restrict__ mask, int M, float w,
                      float* __restrict__ out) {
  int wv = blockIdx.x * 8 + threadIdx.y;
  int lane = threadIdx.x;
  if (wv >= M) return;
  int node = mask[wv];
  float dd = 0.0f, xx = 0.0f, yy = 0.0f;
#pragma unroll
  for (int j = 0; j < 8; j++) {
    size_t idx = (size_t)node * OUT_H + j * 32 + lane;
    float x = X[idx], y = Yc[idx];
    dd += x * y; xx += x * x; yy += y * y;
  }
  for (int o = 16; o > 0; o >>= 1) {
    dd += __shfl_xor(dd, o, 32);
    xx += __shfl_xor(xx, o, 32);
    yy += __shfl_xor(yy, o, 32);
  }
  if (lane == 0) {
    float nx = fmaxf(sqrtf(xx), NORM_EPS);
    float ny = fmaxf(sqrtf(yy), NORM_EPS);
    float cs = dd / (nx * ny);
    atomicAdd(out, w * (1.0f - cs) / (float)M);
  }
}

// ---------------- host orchestration ----------------

extern "C" void kernel_launch(void* const* d_in, const int* in_sizes, int n_in,
                              void* d_out, int out_size, void* d_ws, size_t ws_size,
                              hipStream_t stream) {
  const float* feat  = (const float*)d_in[0];
  const int*   src   = (const int*)  d_in[1];
  const int*   dst   = (const int*)  d_in[2];
  const int*   mask0 = (const int*)  d_in[3];
  const int*   mask1 = (const int*)  d_in[4];
  const float* tok0  = (const float*)d_in[5];
  const float* tok1  = (const float*)d_in[6];
  const float* encW0 = (const float*)d_in[7];
  const float* encB0 = (const float*)d_in[8];
  const float* encCa0= (const float*)d_in[9];
  const float* encG0 = (const float*)d_in[10];
  const float* encBe0= (const float*)d_in[11];
  const float* encAa0= (const float*)d_in[12];
  const float* encW1 = (const float*)d_in[13];
  const float* encB1 = (const float*)d_in[14];
  const float* encCa1= (const float*)d_in[15];
  const float* encG1 = (const float*)d_in[16];
  const float* encBe1= (const float*)d_in[17];
  const float* encAa1= (const float*)d_in[18];
  const float* decW0 = (const float*)d_in[19];
  const float* decB0 = (const float*)d_in[20];
  const float* decCa0= (const float*)d_in[21];
  const float* decG0 = (const float*)d_in[22];
  const float* decBe0= (const float*)d_in[23];
  const float* decAa0= (const float*)d_in[24];

  const int N  = in_sizes[0] / IN_H;
  const int E  = in_sizes[1];
  const int M0 = in_sizes[3];
  const int M1 = in_sizes[4];

  // workspace carve-out (256B aligned slices)
  char* p = (char*)d_ws;
  auto carve = [&](size_t bytes) -> void* {
    void* r = (void*)p;
    p += (bytes + 255) & ~(size_t)255;
    return r;
  };
  float* nsrc = (float*)carve((size_t)N * 4);
  float* ndst = (float*)carve((size_t)N * 4);
  int*   m0   = (int*)  carve((size_t)N * 4);
  int*   m1   = (int*)  carve((size_t)N * 4);
  float* bnS  = (float*)carve(OUT_H * 4);
  float* bnQ  = (float*)carve(OUT_H * 4);
  float* aggX  = (float*)carve((size_t)N * IN_H * 4);   // contiguous trio;
  float* aggF  = (float*)carve((size_t)N * IN_H * 4);   // aggX..aggF reused as agg256
  float* aggX1 = (float*)carve((size_t)N * IN_H * 4);
  float* H0 = (float*)carve((size_t)N * OUT_H * 4);     // enc-b1 L0 out -> h1
  float* H1 = (float*)carve((size_t)N * OUT_H * 4);     // enc-b2 L0 out -> h1b
  float* H2 = (float*)carve((size_t)N * OUT_H * 4);     // dec(feat) out
  float* H3 = (float*)carve((size_t)N * OUT_H * 4);     // dec(x1) out
  float* agg256 = aggX;                                 // overlay after stage-D GEMMs

  // zero init
  hipMemsetAsync(nsrc, 0, (size_t)N * 4, stream);
  hipMemsetAsync(ndst, 0, (size_t)N * 4, stream);
  hipMemsetAsync(m0,   0, (size_t)N * 4, stream);
  hipMemsetAsync(m1,   0, (size_t)N * 4, stream);
  hipMemsetAsync(aggX, 0, (size_t)N * IN_H * 4 * 3, stream);
  hipMemsetAsync(d_out, 0, sizeof(float), stream);

  dim3 b256(256);
  k_deg<<<(E + 255) / 256, b256, 0, stream>>>(src, dst, nsrc, ndst, E);
  k_norm_inplace<<<(N + 255) / 256, b256, 0, stream>>>(nsrc, ndst, N);
  k_mark<<<(M0 + 255) / 256, b256, 0, stream>>>(mask0, M0, m0);
  k_mark<<<(M1 + 255) / 256, b256, 0, stream>>>(mask1, M1, m1);

  {
    long long tt = (long long)E * 32;
    k_spmm128_fused<<<(unsigned)((tt + 255) / 256), b256, 0, stream>>>(
        (const float4*)feat, (const float4*)tok0, (const float4*)tok1,
        src, dst, nsrc, m0, m1, aggF, aggX, aggX1, E);
  }

  dim3 gb(32, 8);
  dim3 gg((N + 15) / 16, 1);                    // each wave: 16x32 strip, 8 waves = 256 cols
  auto gemm_bn = [&](const float* A, int K, const float* W, const float* bi,
                     const float* ca, const float* g, const float* be,
                     const float* aa, float* Y) {
    k_gemm_bias_prelu<<<gg, gb, 0, stream>>>(A, ndst, W, bi, ca, Y, N, K);
    hipMemsetAsync(bnS, 0, OUT_H * 4, stream);
    hipMemsetAsync(bnQ, 0, OUT_H * 4, stream);
    k_bn_stats<<<(N + 63) / 64, OUT_H, 0, stream>>>(Y, bnS, bnQ, N);
    k_bn_apply<<<(unsigned)(((size_t)N * OUT_H + 255) / 256), b256, 0, stream>>>(
        Y, bnS, bnQ, g, be, aa, N);
  };

  // layer-0 of all four paths (K = 128)
  gemm_bn(aggX,  IN_H, encW0, encB0, encCa0, encG0, encBe0, encAa0, H0);
  gemm_bn(aggF,  IN_H, encW0, encB0, encCa0, encG0, encBe0, encAa0, H1);
  gemm_bn(aggF,  IN_H, decW0, decB0, decCa0, decG0, decBe0, decAa0, H2);
  gemm_bn(aggX1, IN_H, decW0, decB0, decCa0, decG0, decBe0, decAa0, H3);

  long long tt = (long long)E * 32;
  // encoder layer-1, branch 1 (h1 overwrites H0)
  hipMemsetAsync(agg256, 0, (size_t)N * OUT_H * 4, stream);
  k_spmm256<<<(unsigned)((tt + 255) / 256), b256, 0, stream>>>(
      (const float4*)H0, src, dst, nsrc, agg256, E);
  gemm_bn(agg256, OUT_H, encW1, encB1, encCa1, encG1, encBe1, encAa1, H0);

  // encoder layer-1, branch 2 (h1b overwrites H1)
  hipMemsetAsync(agg256, 0, (size_t)N * OUT_H * 4, stream);
  k_spmm256<<<(unsigned)((tt + 255) / 256), b256, 0, stream>>>(
      (const float4*)H1, src, dst, nsrc, agg256, E);
  gemm_bn(agg256, OUT_H, encW1, encB1, encCa1, encG1, encBe1, encAa1, H1);

  // losses: 0.5*sce(h1[mask0], dec_feat[mask0]) + 0.5*sce(h1b[mask1], dec_x1[mask1])
  k_sce<<<(M0 + 7) / 8, dim3(32, 8), 0, stream>>>(H0, H2, mask0, M0, 0.5f, (float*)d_out);
  k_sce<<<(M1 + 7) / 8, dim3(32, 8), 0, stream>>>(H1, H3, mask1, M1, 0.5f, (float*)d_out);
}